// OrthogonalButterfly_6390911337182
// MI455X (gfx1250) — compile-verified
//
#include <hip/hip_runtime.h>
#include <math.h>

#define WIDTH 4096
#define HALF  2048
#define DEPTH 12
#define BITS  12
#define TPB   512
#define ROWS_PER_BLOCK 4
#define GROUPS (HALF / 2)      /* 1024 radix-4 groups per row       */
#define GPT    (GROUPS / TPB)  /* 2 groups per thread               */

// ---------- CDNA5 async-to-LDS detection ----------
#if defined(__has_builtin)
# if __has_builtin(__builtin_amdgcn_global_load_async_to_lds_b128)
#  define HAVE_ASYNC_LDS 1
# endif
# if __has_builtin(__builtin_amdgcn_s_wait_asynccnt)
#  define HAVE_WAIT_ASYNC 1
# endif
#endif
#ifndef HAVE_ASYNC_LDS
# define HAVE_ASYNC_LDS 0
#endif
#ifndef HAVE_WAIT_ASYNC
# define HAVE_WAIT_ASYNC 0
#endif

typedef int v4i __attribute__((ext_vector_type(4)));
typedef __attribute__((address_space(1))) v4i global_v4i;
typedef __attribute__((address_space(3))) v4i lds_v4i;

__device__ __forceinline__ void wait_async_zero() {
#if HAVE_WAIT_ASYNC
  __builtin_amdgcn_s_wait_asynccnt(0);
#else
  asm volatile("s_wait_asynccnt 0" ::: "memory");
#endif
}

// angle index for pair-slot q at step d: insert 0-bit at (11-d), rotl by d
__device__ __forceinline__ int angle_index(int d, int q) {
  int sh = (BITS - 1) - d;
  int stride = 1 << sh;
  int p = ((q >> sh) << (sh + 1)) | (q & (stride - 1));
  return ((p << d) | (p >> (BITS - d))) & (WIDTH - 1);
}

__device__ __forceinline__ float2 cs_inline(const float* angles, int d, int q) {
  int j = angle_index(d, q);
  float sv, cv;
  sincosf(angles[(size_t)j * DEPTH + d], &sv, &cv);
  return make_float2(cv, sv);
}

// ---------- pre-kernel: pre-permuted (cos,sin) table, linear in (d, q) ----------
__global__ void cs_precompute(const float* __restrict__ angles, float2* __restrict__ cs) {
  int idx = blockIdx.x * blockDim.x + threadIdx.x;
  if (idx >= DEPTH * HALF) return;
  int d = idx / HALF;
  int q = idx - d * HALF;
  int j = angle_index(d, q);
  float sv, cv;
  sincosf(angles[(size_t)j * DEPTH + d], &sv, &cv);
  cs[idx] = make_float2(cv, sv);
}

// ---------- main fused kernel: 12 steps in 6 radix-4 rounds ----------
template <bool USE_TABLE>
__global__ __launch_bounds__(TPB) void butterfly_main(const float* __restrict__ X,
                                                      const float* __restrict__ angles,
                                                      const float2* __restrict__ cs,
                                                      float* __restrict__ out,
                                                      int nRows) {
  __shared__ __align__(16) float lds[ROWS_PER_BLOCK][WIDTH];
  const int t = threadIdx.x;
  const long rowBase = (long)blockIdx.x * ROWS_PER_BLOCK;

  // ---- stage rows into LDS ----
#if HAVE_ASYNC_LDS
  for (int r = 0; r < ROWS_PER_BLOCK; ++r) {
    if (rowBase + r < nRows) {
      const float* gp = X + (rowBase + r) * (long)WIDTH;
#pragma unroll
      for (int k = 0; k < WIDTH / (4 * TPB); ++k) {  // 2 x b128 per thread per row
        int e = (k * TPB + t) * 4;
        __builtin_amdgcn_global_load_async_to_lds_b128(
            (global_v4i*)(gp + e), (lds_v4i*)&lds[r][e], 0, 0);
      }
    }
  }
  wait_async_zero();
  __syncthreads();
#else
  for (int r = 0; r < ROWS_PER_BLOCK; ++r) {
    if (rowBase + r < nRows) {
      const float4* gp = (const float4*)(X + (rowBase + r) * (long)WIDTH);
#pragma unroll
      for (int k = 0; k < WIDTH / (4 * TPB); ++k) {
        int e = k * TPB + t;
        *(float4*)&lds[r][e * 4] = gp[e];
      }
    }
  }
  __syncthreads();
#endif

  // ---- 6 rounds, 2 butterfly steps per round (radix-4) ----
#pragma unroll
  for (int dd = 0; dd < DEPTH / 2; ++dd) {
    const int d = 2 * dd;
    const int sh = (BITS - 1) - d;  // 11,9,7,5,3,1
    const int s = 1 << sh;
    const int s2 = s >> 1;

    float2 ca[GPT], cb[GPT], cc[GPT], cd[GPT];
    int pbase[GPT];
#pragma unroll
    for (int k = 0; k < GPT; ++k) {
      int g = k * TPB + t;          // group id, 0..1023
      int low = g & (s2 - 1);
      int hi = g >> (sh - 1);
      int p = (hi << (sh + 1)) | low;   // two zero bits inserted at sh, sh-1
      int q1 = (hi << sh) | low;        // pair slot for (p, p+s) at step d
      int q2 = q1 + s2;                 // pair slot for (p+s2, p+s+s2)
      pbase[k] = p;
      if (USE_TABLE) {
        ca[k] = cs[(size_t)d * HALF + q1];
        cb[k] = cs[(size_t)d * HALF + q2];
        cc[k] = cs[(size_t)(d + 1) * HALF + q1];
        cd[k] = cs[(size_t)(d + 1) * HALF + q2];
      } else {
        ca[k] = cs_inline(angles, d, q1);
        cb[k] = cs_inline(angles, d, q2);
        cc[k] = cs_inline(angles, d + 1, q1);
        cd[k] = cs_inline(angles, d + 1, q2);
      }
    }

#pragma unroll
    for (int r = 0; r < ROWS_PER_BLOCK; ++r) {
      float* row = lds[r];
#pragma unroll
      for (int k = 0; k < GPT; ++k) {
        int p = pbase[k];
        float x0 = row[p];
        float x1 = row[p + s2];
        float x2 = row[p + s];
        float x3 = row[p + s + s2];
        // step d (stride s): pairs (x0,x2) and (x1,x3)
        float y0 = fmaf(x0, ca[k].x, x2 * ca[k].y);
        float y2 = fmaf(x2, ca[k].x, -(x0 * ca[k].y));
        float y1 = fmaf(x1, cb[k].x, x3 * cb[k].y);
        float y3 = fmaf(x3, cb[k].x, -(x1 * cb[k].y));
        // step d+1 (stride s2): pairs (y0,y1) and (y2,y3)
        float z0 = fmaf(y0, cc[k].x, y1 * cc[k].y);
        float z1 = fmaf(y1, cc[k].x, -(y0 * cc[k].y));
        float z2 = fmaf(y2, cd[k].x, y3 * cd[k].y);
        float z3 = fmaf(y3, cd[k].x, -(y2 * cd[k].y));

        if (dd == DEPTH / 2 - 1) {
          // last round: s=2, s2=1 -> p..p+3 contiguous; store straight to global
          if (rowBase + r < nRows) {
            *(float4*)(out + (rowBase + r) * (long)WIDTH + p) =
                make_float4(z0, z1, z2, z3);
          }
        } else {
          row[p] = z0;
          row[p + s2] = z1;
          row[p + s] = z2;
          row[p + s + s2] = z3;
        }
      }
    }
    if (dd != DEPTH / 2 - 1) __syncthreads();
  }
}

extern "C" void kernel_launch(void* const* d_in, const int* in_sizes, int n_in,
                              void* d_out, int out_size, void* d_ws, size_t ws_size,
                              hipStream_t stream) {
  const float* X = (const float*)d_in[0];
  const float* angles = (const float*)d_in[1];
  float* out = (float*)d_out;

  const int nRows = in_sizes[0] / WIDTH;  // 8192
  const int nBlocks = (nRows + ROWS_PER_BLOCK - 1) / ROWS_PER_BLOCK;
  const size_t csBytes = (size_t)DEPTH * HALF * sizeof(float2);  // 196608 B

  if (d_ws != nullptr && ws_size >= csBytes) {
    float2* cs = (float2*)d_ws;
    cs_precompute<<<(DEPTH * HALF + 255) / 256, 256, 0, stream>>>(angles, cs);
    butterfly_main<true><<<nBlocks, TPB, 0, stream>>>(X, angles, cs, out, nRows);
  } else {
    butterfly_main<false><<<nBlocks, TPB, 0, stream>>>(X, angles, nullptr, out, nRows);
  }
}